// BaseRouter_26242250178691
// MI455X (gfx1250) — compile-verified
//
#include <hip/hip_runtime.h>
#include <hip/hip_bf16.h>

#define NUM_TOKENS  16384
#define D_MODEL     2048
#define NUM_EXPERTS 64
#define TOP_K       2
#define CAPACITY    640
#define FLATN       (NUM_TOKENS * TOP_K)   // 32768, indices fit in 15 bits

typedef __attribute__((ext_vector_type(2))) float v2f;
typedef __attribute__((ext_vector_type(8))) float v8f;

// ---------------------------------------------------------------------------
// Kernel 1: fused router GEMM (fp32 WMMA 16x16x4) + bias + softmax + top-2.
// One wave computes a 32(token) x 64(expert) logits block:
//   per K-step: 2 A-tiles (16x4) x 4 B-tiles (4x16) -> 8 v_wmma_f32_16x16x4_f32,
//   B registers reused across both A tiles (halves W traffic from L2).
// Logits never leave the WGP: parked in LDS, softmax/top-2 reduced in place.
// ---------------------------------------------------------------------------
__global__ __launch_bounds__(256) void router_fused_kernel(
    const float* __restrict__ x,     // [T, D] row-major
    const float* __restrict__ W,     // [E, D] row-major
    const float* __restrict__ bias,  // [E]
    float* __restrict__ unnorm,      // [T*2] unnormalized top-k probs
    float* __restrict__ normp,       // [T*2] normalized top-k probs
    int*   __restrict__ eid)         // [T*2] chosen expert ids
{
    // per-wave 16x64 logits tile; row stride 68 dwords => (4*lane+e)%64 banks, conflict-free
    __shared__ float lds[8][16][68];

    const int wave = threadIdx.x >> 5;
    const int lane = threadIdx.x & 31;
    const int half = lane >> 4;      // 0 -> holds {K0,K1}; 1 -> holds {K2,K3} of each K-step
    const int l16  = lane & 15;

    const int unit = blockIdx.x * 8 + wave;   // 512 wave units total
    const int m0   = unit * 32;               // first of this wave's 32 tokens

    // A 16x4 f32 layout: lane L<16 = row M=L {K0,K1}; lane L>=16 = row M=L-16 {K2,K3}
    const float* xrow0 = x + (size_t)(m0 + l16) * D_MODEL + 2 * half;
    const float* xrow1 = xrow0 + (size_t)16 * D_MODEL;
    // B 4x16 f32 layout (K x N), B = W^T: lane L<16 = col N=L {K0,K1}; L>=16 {K2,K3}
    const float* wp0 = W + (size_t)l16 * D_MODEL + 2 * half;
    const float* wp1 = wp0 + (size_t)16 * D_MODEL;
    const float* wp2 = wp0 + (size_t)32 * D_MODEL;
    const float* wp3 = wp0 + (size_t)48 * D_MODEL;

    v8f acc[2][4];
    v8f zero = {};
#pragma unroll
    for (int m = 0; m < 2; ++m)
#pragma unroll
        for (int t = 0; t < 4; ++t) acc[m][t] = zero;

    for (int kb = 0; kb < D_MODEL; kb += 4) {
        v2f a0 = *(const v2f*)(xrow0 + kb);
        v2f a1 = *(const v2f*)(xrow1 + kb);
        v2f b0 = *(const v2f*)(wp0 + kb);
        v2f b1 = *(const v2f*)(wp1 + kb);
        v2f b2 = *(const v2f*)(wp2 + kb);
        v2f b3 = *(const v2f*)(wp3 + kb);
        acc[0][0] = __builtin_amdgcn_wmma_f32_16x16x4_f32(false, a0, false, b0, (short)0, acc[0][0], false, false);
        acc[1][0] = __builtin_amdgcn_wmma_f32_16x16x4_f32(false, a1, false, b0, (short)0, acc[1][0], false, false);
        acc[0][1] = __builtin_amdgcn_wmma_f32_16x16x4_f32(false, a0, false, b1, (short)0, acc[0][1], false, false);
        acc[1][1] = __builtin_amdgcn_wmma_f32_16x16x4_f32(false, a1, false, b1, (short)0, acc[1][1], false, false);
        acc[0][2] = __builtin_amdgcn_wmma_f32_16x16x4_f32(false, a0, false, b2, (short)0, acc[0][2], false, false);
        acc[1][2] = __builtin_amdgcn_wmma_f32_16x16x4_f32(false, a1, false, b2, (short)0, acc[1][2], false, false);
        acc[0][3] = __builtin_amdgcn_wmma_f32_16x16x4_f32(false, a0, false, b3, (short)0, acc[0][3], false, false);
        acc[1][3] = __builtin_amdgcn_wmma_f32_16x16x4_f32(false, a1, false, b3, (short)0, acc[1][3], false, false);
    }

    float bsv[4];
#pragma unroll
    for (int t = 0; t < 4; ++t) bsv[t] = bias[t * 16 + l16];

    // Two 16-token sub-tiles share the same LDS staging area sequentially.
#pragma unroll
    for (int m = 0; m < 2; ++m) {
        // C/D layout: VGPR v, lanes 0-15 -> M=v; lanes 16-31 -> M=v+8; N = lane&15 within tile t
#pragma unroll
        for (int r = 0; r < 8; ++r) {
#pragma unroll
            for (int t = 0; t < 4; ++t)
                lds[wave][half * 8 + r][t * 16 + l16] = acc[m][t][r] + bsv[t];
        }
        __syncthreads();
        if (lane < 16) {
            const float* lrow = lds[wave][lane];
            float mx = -3.402823466e38f;
#pragma unroll 8
            for (int e = 0; e < NUM_EXPERTS; ++e) mx = fmaxf(mx, lrow[e]);
            float sum = 0.0f;
            float b1v = -3.402823466e38f, b2v = -3.402823466e38f;
            int   i1 = 0, i2 = 0;
            for (int e = 0; e < NUM_EXPERTS; ++e) {
                float lv = lrow[e];
                sum += __expf(lv - mx);
                if (lv > b1v)      { b2v = b1v; i2 = i1; b1v = lv; i1 = e; }  // ties -> lowest index
                else if (lv > b2v) { b2v = lv; i2 = e; }
            }
            float rs  = 1.0f / sum;
            float p1  = __expf(b1v - mx) * rs;
            float p2  = __expf(b2v - mx) * rs;
            float inv = 1.0f / (p1 + p2);
            int tok = m0 + m * 16 + lane;
            unnorm[2 * tok]     = p1;
            unnorm[2 * tok + 1] = p2;
            normp[2 * tok]      = p1 * inv;
            normp[2 * tok + 1]  = p2 * inv;
            eid[2 * tok]        = i1;
            eid[2 * tok + 1]    = i2;
        }
        __syncthreads();
    }
}

// ---------------------------------------------------------------------------
// Kernel 2: per-expert capacity selection (one workgroup per expert).
// Compact this expert's candidates into LDS as orderable 64-bit keys
// (softmax probs > 0 so IEEE bits order monotonically; low 15 bits encode
// inverted flat index so equal probs break ties toward lower flat index,
// matching jax.lax.top_k stability). Then O(n^2) rank via LDS broadcast reads.
// Worst case 32768 keys * 8B = 256KB dynamic LDS (< 320KB/WGP on CDNA5).
// ---------------------------------------------------------------------------
__global__ __launch_bounds__(256) void expert_capacity_kernel(
    const float* __restrict__ unnorm,
    const float* __restrict__ normp,
    const int*   __restrict__ eid,
    float* __restrict__ oprob,   // [E, CAP]
    int*   __restrict__ oidx)    // [E, CAP]
{
    extern __shared__ unsigned long long keys[];
    __shared__ int cnt;

    const int e   = blockIdx.x;
    const int tid = threadIdx.x;

    for (int s = tid; s < CAPACITY; s += 256) {
        oprob[e * CAPACITY + s] = 0.0f;
        oidx[e * CAPACITY + s]  = -1;
    }
    if (tid == 0) cnt = 0;
    __syncthreads();

    for (int i = tid; i < FLATN; i += 256) {
        if (eid[i] == e) {
            unsigned ub = __float_as_uint(unnorm[i]);
            int p = atomicAdd(&cnt, 1);
            keys[p] = ((unsigned long long)ub << 15)
                    | (unsigned long long)(unsigned)(FLATN - 1 - i);
        }
    }
    __syncthreads();

    const int n = cnt;
    for (int c = tid; c < n; c += 256) {
        unsigned long long kc = keys[c];
        int rank = 0;
#pragma unroll 8
        for (int j = 0; j < n; ++j)
            rank += (keys[j] > kc) ? 1 : 0;   // same-address LDS read -> broadcast
        if (rank < CAPACITY) {
            int i = FLATN - 1 - (int)(kc & 0x7FFFULL);
            oprob[e * CAPACITY + rank] = normp[i];
            oidx[e * CAPACITY + rank]  = i >> 1;   // flat = token*TOP_K + slot
        }
    }
}

extern "C" void kernel_launch(void* const* d_in, const int* in_sizes, int n_in,
                              void* d_out, int out_size, void* d_ws, size_t ws_size,
                              hipStream_t stream) {
    (void)in_sizes; (void)n_in; (void)out_size; (void)ws_size;
    const float* x  = (const float*)d_in[0];
    const float* W  = (const float*)d_in[1];
    const float* b  = (const float*)d_in[2];
    // d_in[3] = top_k (2), d_in[4] = expert_capacity (640): compile-time constants here.

    float* unnorm = (float*)d_ws;               // [FLATN]
    float* normp  = unnorm + FLATN;             // [FLATN]
    int*   eid    = (int*)(normp + FLATN);      // [FLATN]  (384 KB total scratch)

    float* oprob = (float*)d_out;                                   // [E, CAP] f32
    int*   oidx  = (int*)((float*)d_out + NUM_EXPERTS * CAPACITY);  // [E, CAP] i32

    // 64 blocks x 256 threads; each wave handles 32 tokens x 64 experts.
    router_fused_kernel<<<NUM_TOKENS / 256, 256, 0, stream>>>(x, W, b, unnorm, normp, eid);

    size_t smem = (size_t)FLATN * sizeof(unsigned long long);       // 256 KB
    (void)hipFuncSetAttribute((const void*)expert_capacity_kernel,
                              hipFuncAttributeMaxDynamicSharedMemorySize, (int)smem);
    expert_capacity_kernel<<<NUM_EXPERTS, 256, smem, stream>>>(unnorm, normp, eid, oprob, oidx);
}